// MHAttention_67164698575169
// MI455X (gfx1250) — compile-verified
//
#include <hip/hip_runtime.h>
#include <hip/hip_bf16.h>

typedef float v2f __attribute__((ext_vector_type(2)));
typedef float v8f __attribute__((ext_vector_type(8)));

#define B_  4
#define T_  2048
#define D_  1024
#define H_  16
#define HD_ 64

// ---------------------------------------------------------------------------
// fp32 GEMM: C[M,N] = A[M,K] @ B[K,N] + bias[N]
// Block tile 128x256, BK=32, 256 threads = 8 waves (2 M x 4 N),
// wave tile 64x64 = 16 accumulator fragments -> 16 WMMAs per 8 LDS loads.
// Global->LDS staging is register double-buffered: next chunk's loads are in
// flight while the current chunk computes, so the matrix pipe never waits on
// HBM/L2 latency.
// ---------------------------------------------------------------------------
__global__ __launch_bounds__(256) void gemm_wmma_f32(
    const float* __restrict__ A, const float* __restrict__ Bm,
    const float* __restrict__ bias, float* __restrict__ C,
    int M, int N, int K) {
  __shared__ float Al[128][34];  // [m][k], +2 pad vs 64 banks
  __shared__ float Bl[256][34];  // transposed: [n][k]

  const int tid  = threadIdx.x;
  const int wave = tid >> 5;
  const int lane = tid & 31;
  const int half = lane >> 4;
  const int l16  = lane & 15;
  const int wm   = wave >> 2;    // 0..1  (M direction, 64 rows each)
  const int wn   = wave & 3;     // 0..3  (N direction, 64 cols each)
  const int gm0  = blockIdx.y * 128;
  const int gn0  = blockIdx.x * 256;

  v8f acc[4][4] = {};

  float4 abuf[4], bbuf[8];

  // --- issue loads for chunk 0 ---
#pragma unroll
  for (int i = 0; i < 4; ++i) {
    const int f   = tid + i * 256;
    const int row = f >> 3;
    const int c4  = (f & 7) << 2;
    abuf[i] = *reinterpret_cast<const float4*>(&A[(size_t)(gm0 + row) * K + c4]);
  }
#pragma unroll
  for (int i = 0; i < 8; ++i) {
    const int f   = tid + i * 256;
    const int row = f >> 6;
    const int c4  = (f & 63) << 2;
    bbuf[i] = *reinterpret_cast<const float4*>(&Bm[(size_t)row * N + gn0 + c4]);
  }

  for (int k0 = 0; k0 < K; k0 += 32) {
    // --- dump staged registers to LDS ---
#pragma unroll
    for (int i = 0; i < 4; ++i) {
      const int f   = tid + i * 256;
      const int row = f >> 3;
      const int c4  = (f & 7) << 2;
      Al[row][c4 + 0] = abuf[i].x; Al[row][c4 + 1] = abuf[i].y;
      Al[row][c4 + 2] = abuf[i].z; Al[row][c4 + 3] = abuf[i].w;
    }
#pragma unroll
    for (int i = 0; i < 8; ++i) {
      const int f   = tid + i * 256;
      const int row = f >> 6;
      const int c4  = (f & 63) << 2;
      Bl[c4 + 0][row] = bbuf[i].x; Bl[c4 + 1][row] = bbuf[i].y;
      Bl[c4 + 2][row] = bbuf[i].z; Bl[c4 + 3][row] = bbuf[i].w;
    }
    __syncthreads();

    // --- issue next chunk's global loads; results consumed next iteration ---
    const int kn = k0 + 32;
    if (kn < K) {
#pragma unroll
      for (int i = 0; i < 4; ++i) {
        const int f   = tid + i * 256;
        const int row = f >> 3;
        const int c4  = (f & 7) << 2;
        const float* src = &A[(size_t)(gm0 + row) * K + kn + c4];
        abuf[i] = *reinterpret_cast<const float4*>(src);
        __builtin_prefetch(src + 32, 0, 1);   // chunk k0+64 -> global_prefetch_b8
      }
#pragma unroll
      for (int i = 0; i < 8; ++i) {
        const int f   = tid + i * 256;
        const int row = f >> 6;
        const int c4  = (f & 63) << 2;
        const float* src = &Bm[(size_t)(kn + row) * N + gn0 + c4];
        bbuf[i] = *reinterpret_cast<const float4*>(src);
        __builtin_prefetch(src + 32 * N, 0, 1);
      }
    }

    // --- compute: 8 K-steps x 16 WMMAs ---
#pragma unroll
    for (int kk = 0; kk < 32; kk += 4) {
      const int kidx = kk + half * 2;  // lanes 0-15: k,k+1 ; lanes 16-31: k+2,k+3
      v2f af[4], bf[4];
#pragma unroll
      for (int mi = 0; mi < 4; ++mi) {
        const int row = wm * 64 + mi * 16 + l16;
        af[mi] = *reinterpret_cast<const v2f*>(&Al[row][kidx]);
      }
#pragma unroll
      for (int ni = 0; ni < 4; ++ni) {
        const int col = wn * 64 + ni * 16 + l16;
        bf[ni] = *reinterpret_cast<const v2f*>(&Bl[col][kidx]);
      }
#pragma unroll
      for (int mi = 0; mi < 4; ++mi)
#pragma unroll
        for (int ni = 0; ni < 4; ++ni)
          acc[mi][ni] = __builtin_amdgcn_wmma_f32_16x16x4_f32(
              false, af[mi], false, bf[ni], (short)0, acc[mi][ni], false, false);
    }
    __syncthreads();
  }

  // --- epilogue: C-layout -> global, add bias ---
#pragma unroll
  for (int ni = 0; ni < 4; ++ni) {
    const int col = gn0 + wn * 64 + ni * 16 + l16;
    const float bv = bias[col];
#pragma unroll
    for (int mi = 0; mi < 4; ++mi) {
#pragma unroll
      for (int r = 0; r < 8; ++r) {
        const int row = gm0 + wm * 64 + mi * 16 + half * 8 + r;
        C[(size_t)row * N + col] = acc[mi][ni][r] + bv;
      }
    }
  }
}

// ---------------------------------------------------------------------------
// Flash attention: one wave per (b, h, 16-row q tile).
// qkv layout [B, T, 3D]; output [B, T, D] (heads re-interleaved).
// K/V fragments are batch-loaded into registers per key tile so global
// latency overlaps the WMMA chains and the softmax VALU work.
// ---------------------------------------------------------------------------
__global__ __launch_bounds__(32) void attn_wmma(
    const float* __restrict__ qkv, float* __restrict__ out) {
  __shared__ float Pl[16][18];  // P tile staging (C-layout -> A-layout)

  const int lane = threadIdx.x;
  const int half = lane >> 4;
  const int l16  = lane & 15;

  const int QT = T_ / 16;
  int idx = blockIdx.x;
  const int qi = idx % QT; idx /= QT;
  const int h  = idx % H_;
  const int b  = idx / H_;
  const int q0 = qi * 16;

  const size_t rs = (size_t)3 * D_;                 // row stride in qkv
  const float* Qb = qkv + (size_t)b * T_ * rs + h * HD_;
  const float* Kb = Qb + D_;
  const float* Vb = Qb + 2 * D_;

  // Q fragments, register resident: lane<16 holds row l16 K={4k,4k+1},
  // lane>=16 holds row l16 K={4k+2,4k+3}
  float qa[32];
#pragma unroll
  for (int k4 = 0; k4 < 16; ++k4) {
    const int kidx = k4 * 4 + half * 2;
    const float2 v = *reinterpret_cast<const float2*>(
        &Qb[(size_t)(q0 + l16) * rs + kidx]);
    qa[2 * k4] = v.x; qa[2 * k4 + 1] = v.y;
  }

  float mrow[8], lrow[8];
  v8f oacc[4] = {};
#pragma unroll
  for (int r = 0; r < 8; ++r) { mrow[r] = -3.0e38f; lrow[r] = 0.0f; }

  for (int j = 0; j <= qi; ++j) {
    const int k0 = j * 16;

    // ---- batch-load K fragments (16 x b64) ----
    float2 kf[16];
#pragma unroll
    for (int k4 = 0; k4 < 16; ++k4) {
      const int kidx = k4 * 4 + half * 2;
      kf[k4] = *reinterpret_cast<const float2*>(
          &Kb[(size_t)(k0 + l16) * rs + kidx]);
    }
    // ---- batch-load V fragments (overlap with QK^T + softmax) ----
    v2f vfr[4][4];
#pragma unroll
    for (int kk4 = 0; kk4 < 4; ++kk4) {
      const int kidx = kk4 * 4 + half * 2;
#pragma unroll
      for (int ni = 0; ni < 4; ++ni) {
        const int n = ni * 16 + l16;
        vfr[kk4][ni][0] = Vb[(size_t)(k0 + kidx) * rs + n];
        vfr[kk4][ni][1] = Vb[(size_t)(k0 + kidx + 1) * rs + n];
      }
    }

    // ---- S = Q @ K^T  (16 fp32 WMMAs, K-dim 64) ----
    v8f s = {};
#pragma unroll
    for (int k4 = 0; k4 < 16; ++k4) {
      v2f a; a[0] = qa[2 * k4]; a[1] = qa[2 * k4 + 1];
      v2f bf; bf[0] = kf[k4].x; bf[1] = kf[k4].y;
      s = __builtin_amdgcn_wmma_f32_16x16x4_f32(
          false, a, false, bf, (short)0, s, false, false);
    }

    // ---- online softmax (rows live across 16-lane halves) ----
    const bool diag = (j == qi);
    float p[8], corr[8];
#pragma unroll
    for (int r = 0; r < 8; ++r) {
      float sv = s[r] * 0.125f;                   // 1/sqrt(64)
      const int rowm = r + half * 8;
      if (diag && (l16 > rowm)) sv = -3.0e38f;    // causal mask
      float mx = sv;
      mx = fmaxf(mx, __shfl_xor(mx, 1, 32));
      mx = fmaxf(mx, __shfl_xor(mx, 2, 32));
      mx = fmaxf(mx, __shfl_xor(mx, 4, 32));
      mx = fmaxf(mx, __shfl_xor(mx, 8, 32));
      const float mnew = fmaxf(mrow[r], mx);
      const float c  = __expf(mrow[r] - mnew);
      const float pv = __expf(sv - mnew);
      float sum = pv;
      sum += __shfl_xor(sum, 1, 32);
      sum += __shfl_xor(sum, 2, 32);
      sum += __shfl_xor(sum, 4, 32);
      sum += __shfl_xor(sum, 8, 32);
      lrow[r] = lrow[r] * c + sum;
      mrow[r] = mnew;
      corr[r] = c;
      p[r]    = pv;
    }
    // rescale running O by exp(m_old - m_new)
#pragma unroll
    for (int ni = 0; ni < 4; ++ni)
#pragma unroll
      for (int r = 0; r < 8; ++r)
        oacc[ni][r] *= corr[r];

    // ---- stage P through LDS: C-layout -> A-layout ----
#pragma unroll
    for (int r = 0; r < 8; ++r)
      Pl[r + half * 8][l16] = p[r];
    // same-wave DS ops are in-order; compiler inserts s_wait_dscnt for the RAW

    // ---- O += P @ V  (16 fp32 WMMAs) ----
#pragma unroll
    for (int kk4 = 0; kk4 < 4; ++kk4) {
      const int kidx = kk4 * 4 + half * 2;
      const v2f pa = *reinterpret_cast<const v2f*>(&Pl[l16][kidx]);
#pragma unroll
      for (int ni = 0; ni < 4; ++ni)
        oacc[ni] = __builtin_amdgcn_wmma_f32_16x16x4_f32(
            false, pa, false, vfr[kk4][ni], (short)0, oacc[ni], false, false);
    }
  }

  // ---- finalize: O /= l, write [B,T,D] with head interleave ----
  float* Ob = out + ((size_t)b * T_ + q0) * D_ + h * HD_;
#pragma unroll
  for (int ni = 0; ni < 4; ++ni)
#pragma unroll
    for (int r = 0; r < 8; ++r) {
      const int row = r + half * 8;
      Ob[(size_t)row * D_ + ni * 16 + l16] = oacc[ni][r] / lrow[r];
    }
}

// ---------------------------------------------------------------------------
extern "C" void kernel_launch(void* const* d_in, const int* in_sizes, int n_in,
                              void* d_out, int out_size, void* d_ws, size_t ws_size,
                              hipStream_t stream) {
  const float* x      = (const float*)d_in[0];
  const float* qkv_w  = (const float*)d_in[1];
  const float* qkv_b  = (const float*)d_in[2];
  const float* proj_w = (const float*)d_in[3];
  const float* proj_b = (const float*)d_in[4];
  float* out = (float*)d_out;

  float* qkv  = (float*)d_ws;                           // [B,T,3D]  96 MB
  float* attn = qkv + (size_t)B_ * T_ * 3 * D_;         // [B,T,D]   32 MB

  const int M = B_ * T_;

  // 1) qkv = x @ qkv_w + qkv_b   (M=8192, N=3072, K=1024)
  gemm_wmma_f32<<<dim3((3 * D_) / 256, M / 128), dim3(256), 0, stream>>>(
      x, qkv_w, qkv_b, qkv, M, 3 * D_, D_);

  // 2) causal flash attention per (b, h, q-tile)
  attn_wmma<<<dim3(B_ * H_ * (T_ / 16)), dim3(32), 0, stream>>>(qkv, attn);

  // 3) out = attn @ proj_w + proj_b   (M=8192, N=1024, K=1024)
  gemm_wmma_f32<<<dim3(D_ / 256, M / 128), dim3(256), 0, stream>>>(
      attn, proj_w, proj_b, out, M, D_, D_);
}